// NMS_1_74998718922802
// MI455X (gfx1250) — compile-verified
//
#include <hip/hip_runtime.h>

// Greedy per-image NMS for MI455X (gfx1250).
//   boxes : [B, N, 4] f32 (x1,y1,x2,y2)   scores : [B, N] f32
//   out   : [B, KEEP, 4] f32
// One 512-thread workgroup per image; survivors bitmask in LDS; fused
// suppress+argmax passes; 4-deep async global->LDS box pipeline (ASYNCcnt).

#define NBOX    262144
#define NKEEP   20
#define NT      512                    // threads per block (16 wave32s)
#define NWAVES  (NT / 32)              // 16
#define ITEMS   (NBOX / NT)            // 512 boxes per thread
#define MWORDS  (ITEMS / 32)           // 16 mask words per thread
#define BUFS    4                      // async pipeline depth (power of 2)
#define IOU_THR 0.5f
#define EPS_F   1e-9f

#if defined(__gfx1250__) && \
    __has_builtin(__builtin_amdgcn_global_load_async_to_lds_b128) && \
    __has_builtin(__builtin_amdgcn_s_wait_asynccnt)
#define USE_ASYNC_LDS 1
#else
#define USE_ASYNC_LDS 0
#endif

#if USE_ASYNC_LDS
typedef int v4i __attribute__((vector_size(16)));
typedef __attribute__((address_space(1))) v4i* g_v4i_ptr;
typedef __attribute__((address_space(3))) v4i* l_v4i_ptr;

__device__ __forceinline__ void async_box_to_lds(const float4* g, float4* l) {
  // global_load_async_to_lds_b128 (tracked by ASYNCcnt): L2 -> LDS, no VGPR data.
  __builtin_amdgcn_global_load_async_to_lds_b128(
      (g_v4i_ptr)(g),
      (l_v4i_ptr)(l),
      0, 0);
}
#endif

__global__ __launch_bounds__(NT, 1) void nms_gfx1250_kernel(
    const float4* __restrict__ boxes,   // [B*N] float4
    const float*  __restrict__ scores,  // [B*N]
    float4*       __restrict__ out)     // [B*KEEP] float4
{
  const int img  = blockIdx.x;
  const int t    = threadIdx.x;
  const int lane = t & 31;
  const int wave = t >> 5;

  const float4* bb = boxes  + (size_t)img * NBOX;
  const float*  sc = scores + (size_t)img * NBOX;

  __shared__ unsigned s_mask[MWORDS * NT];   // 32 KB survivor bits
  __shared__ float    s_redS[NWAVES];
  __shared__ int      s_redI[NWAVES];
  __shared__ float4   s_win;
#if USE_ASYNC_LDS
  __shared__ __align__(16) float4 s_tile[BUFS][NT];  // 32 KB, 4-deep pipeline
#endif

  // All boxes start valid.
#pragma unroll
  for (int w = 0; w < MWORDS; ++w) s_mask[w * NT + t] = 0xFFFFFFFFu;

  // ---- Round 0: pure argmax over raw scores, vectorized b128 stream ----
  float best = -__builtin_inff();
  int   bi   = 0x7FFFFFFF;
  {
    const float4* sc4 = (const float4*)sc;      // [NBOX/4]
    const int ITEMS4  = NBOX / 4 / NT;          // 128 float4 per thread
#pragma unroll 4
    for (int j = 0; j < ITEMS4; ++j) {
      const int i4 = j * NT + t;
      if (j + 8 < ITEMS4) __builtin_prefetch(sc4 + (j + 8) * NT + t, 0, 0);
      const float4 s4 = sc4[i4];
      const int base = i4 * 4;
      if (s4.x > best) { best = s4.x; bi = base;     }
      if (s4.y > best) { best = s4.y; bi = base + 1; }
      if (s4.z > best) { best = s4.z; bi = base + 2; }
      if (s4.w > best) { best = s4.w; bi = base + 3; }
    }
  }

  for (int k = 0; k < NKEEP; ++k) {
    // ---- block-wide argmax (wave32 shuffle tree, then wave0 over leaders) ----
#pragma unroll
    for (int off = 16; off > 0; off >>= 1) {
      const float os = __shfl_xor(best, off, 32);
      const int   oi = __shfl_xor(bi,   off, 32);
      if (os > best || (os == best && oi < bi)) { best = os; bi = oi; }
    }
    if (lane == 0) { s_redS[wave] = best; s_redI[wave] = bi; }
    __syncthreads();
    if (wave == 0) {
      float b2 = (lane < NWAVES) ? s_redS[lane] : -__builtin_inff();
      int   i2 = (lane < NWAVES) ? s_redI[lane] : 0x7FFFFFFF;
#pragma unroll
      for (int off = 16; off > 0; off >>= 1) {
        const float os = __shfl_xor(b2, off, 32);
        const int   oi = __shfl_xor(i2, off, 32);
        if (os > b2 || (os == b2 && oi < i2)) { b2 = os; i2 = oi; }
      }
      if (lane == 0) {
        const float4 wbox = bb[i2];
        s_win = wbox;
        out[img * NKEEP + k] = wbox;
      }
    }
    __syncthreads();
    if (k == NKEEP - 1) break;

    const float4 wb    = s_win;
    const float  wArea = (wb.z - wb.x) * (wb.w - wb.y);

    // ---- fused pass: suppress vs winner, then argmax over survivors ----
    best = -__builtin_inff();
    bi   = 0x7FFFFFFF;

#if USE_ASYNC_LDS
    // Prime a 4-deep pipeline: tiles 0..2 in flight before the loop.
    async_box_to_lds(bb + 0 * NT + t, &s_tile[0][t]);
    async_box_to_lds(bb + 1 * NT + t, &s_tile[1][t]);
    async_box_to_lds(bb + 2 * NT + t, &s_tile[2][t]);
#endif

#pragma unroll 1
    for (int w = 0; w < MWORDS; ++w) {
      const unsigned mw = s_mask[w * NT + t];
      unsigned       nm = mw;
#pragma unroll 4
      for (int b = 0; b < 32; ++b) {
        const int j   = w * 32 + b;
        const int idx = j * NT + t;
#if USE_ASYNC_LDS
        if (j + 3 < ITEMS) {
          async_box_to_lds(bb + (j + 3) * NT + t, &s_tile[(j + 3) & (BUFS - 1)][t]);
          __builtin_amdgcn_s_wait_asynccnt(3);   // oldest (tile j) landed
        } else if (j + 2 < ITEMS) {
          __builtin_amdgcn_s_wait_asynccnt(2);
        } else if (j + 1 < ITEMS) {
          __builtin_amdgcn_s_wait_asynccnt(1);
        } else {
          __builtin_amdgcn_s_wait_asynccnt(0);
        }
#endif
        if ((mw >> b) & 1u) {
#if USE_ASYNC_LDS
          const float4 box = s_tile[j & (BUFS - 1)][t];  // own slot: no barrier
#else
          if (j + 4 < ITEMS) __builtin_prefetch(bb + (j + 4) * NT + t, 0, 0);
          const float4 box = bb[idx];
#endif
          const float ix1 = fmaxf(wb.x, box.x);
          const float iy1 = fmaxf(wb.y, box.y);
          const float ix2 = fminf(wb.z, box.z);
          const float iy2 = fminf(wb.w, box.w);
          const float inter = fmaxf(ix2 - ix1, 0.0f) * fmaxf(iy2 - iy1, 0.0f);
          const float area  = (box.z - box.x) * (box.w - box.y);
          const float denom = wArea + area - inter + EPS_F;
          if (inter > IOU_THR * denom) {         // iou > 0.5 -> suppress
            nm &= ~(1u << b);
          } else {
            const float s = sc[idx];
            if (s > best || (s == best && idx < bi)) { best = s; bi = idx; }
          }
        }
      }
      if (nm != mw) s_mask[w * NT + t] = nm;
    }
    // no barrier needed here: masks are thread-private, s_redS write is
    // protected by the barrier pair inside the next selection step.
  }
}

extern "C" void kernel_launch(void* const* d_in, const int* in_sizes, int n_in,
                              void* d_out, int out_size, void* d_ws, size_t ws_size,
                              hipStream_t stream) {
  (void)n_in; (void)out_size; (void)d_ws; (void)ws_size;
  const float4* boxes  = (const float4*)d_in[0];
  const float*  scores = (const float*)d_in[1];
  float4*       out    = (float4*)d_out;
  const int B = in_sizes[1] / NBOX;   // scores is [B, N]
  nms_gfx1250_kernel<<<B, NT, 0, stream>>>(boxes, scores, out);
}